// GlobalBlock_30391188586592
// MI455X (gfx1250) — compile-verified
//
#include <hip/hip_runtime.h>

typedef __attribute__((ext_vector_type(2))) float v2f;
typedef __attribute__((ext_vector_type(8))) float v8f;

#define D_DIM 256   // feature dim of edges/nodes
#define H_DIM 256   // hidden dim
#define G_DIM 256   // global dim
#define NB_E 512    // blocks over edge rows
#define NB_N 64     // blocks over node rows
#define THREADS 512 // 16 waves of 32; wave w owns columns [16w, 16w+16)

__device__ __forceinline__ int imin(int a, int b) { return a < b ? a : b; }

// Kernel A: column sums of a row-major [rows x 256] fp32 matrix, expressed as
// ones(16x4) @ B(4x16) WMMA accumulation (C rows replicate the column sum).
// Memory bound (~870 MB total @ 23.3 TB/s): unroll 4 K-tiles (16 rows) per
// iteration so 8 non-temporal dword loads are in flight before the WMMAs
// consume them; two accumulator chains keep the WMMAs independent.
__global__ __launch_bounds__(THREADS)
void colsum_wmma_kernel(const float* __restrict__ edge,
                        const float* __restrict__ node,
                        float* __restrict__ ws, int E, int N) {
    const int lane = threadIdx.x & 31;
    const int wave = threadIdx.x >> 5;
    const int hi   = lane >> 4;              // half-wave: K rows {0,1} vs {2,3}
    const int col  = (wave << 4) + (lane & 15);

    const float* src;
    int rowBeg, rowEnd;
    float* part;
    const int b = blockIdx.x;
    if (b < NB_E) {
        const int chunk = ((E + NB_E - 1) / NB_E + 3) & ~3;
        src    = edge;
        rowBeg = b * chunk;
        rowEnd = imin(rowBeg + chunk, E);
        part   = ws + (long)b * D_DIM;
    } else {
        const int bb = b - NB_E;
        const int chunk = ((N + NB_N - 1) / NB_N + 3) & ~3;
        src    = node;
        rowBeg = bb * chunk;
        rowEnd = imin(rowBeg + chunk, N);
        part   = ws + (long)(NB_E + bb) * D_DIM;
    }

    v8f acc0 = {}, acc1 = {};
    const v2f ones = {1.0f, 1.0f};           // A = all-ones 16x4 (broadcast)

    if (rowBeg < rowEnd) {
        // per-wave base: row (rowBeg + 2*hi), column col
        const float* p = src + (long)(rowBeg + 2 * hi) * D_DIM + col;
        int r = rowBeg;

        // main loop: 16 rows (4 K-tiles) per iteration, 8 loads in flight
        for (; r + 16 <= rowEnd; r += 16, p += 16 * D_DIM) {
            v2f b0, b1, b2, b3;
            b0.x = __builtin_nontemporal_load(p);
            b0.y = __builtin_nontemporal_load(p +      D_DIM);
            b1.x = __builtin_nontemporal_load(p +  4 * D_DIM);
            b1.y = __builtin_nontemporal_load(p +  5 * D_DIM);
            b2.x = __builtin_nontemporal_load(p +  8 * D_DIM);
            b2.y = __builtin_nontemporal_load(p +  9 * D_DIM);
            b3.x = __builtin_nontemporal_load(p + 12 * D_DIM);
            b3.y = __builtin_nontemporal_load(p + 13 * D_DIM);
            acc0 = __builtin_amdgcn_wmma_f32_16x16x4_f32(
                false, ones, false, b0, (short)0, acc0, false, false);
            acc1 = __builtin_amdgcn_wmma_f32_16x16x4_f32(
                false, ones, false, b1, (short)0, acc1, false, false);
            acc0 = __builtin_amdgcn_wmma_f32_16x16x4_f32(
                false, ones, false, b2, (short)0, acc0, false, false);
            acc1 = __builtin_amdgcn_wmma_f32_16x16x4_f32(
                false, ones, false, b3, (short)0, acc1, false, false);
        }
        // remainder: 4 rows at a time
        for (; r + 4 <= rowEnd; r += 4, p += 4 * D_DIM) {
            v2f bm;
            bm.x = __builtin_nontemporal_load(p);
            bm.y = __builtin_nontemporal_load(p + D_DIM);
            acc0 = __builtin_amdgcn_wmma_f32_16x16x4_f32(
                false, ones, false, bm, (short)0, acc0, false, false);
        }
        // ragged tail (<=3 rows), zero-fill OOB lanes
        if (r < rowEnd) {
            const int ra = r + 2 * hi;
            v2f bm;
            bm.x = (ra     < rowEnd) ? src[(long)ra * D_DIM + col]       : 0.0f;
            bm.y = (ra + 1 < rowEnd) ? src[(long)(ra + 1) * D_DIM + col] : 0.0f;
            acc0 = __builtin_amdgcn_wmma_f32_16x16x4_f32(
                false, ones, false, bm, (short)0, acc0, false, false);
        }
    }
    // C row 0 (the column sums) sits in element 0 of lanes 0..15
    if (lane < 16) part[col] = acc0[0] + acc1[0];
}

// Kernel B: reduce partials -> means; then three GEMVs as broadcast-A WMMAs:
//   agg_e = mean_e @ We + be; agg_n = mean_n @ Wn + bn;
//   out   = concat(agg_e, agg_n, g) @ Wu + bu
__global__ __launch_bounds__(THREADS)
void global_update_kernel(const float* __restrict__ ws,
                          const float* __restrict__ global_attr,
                          const float* __restrict__ We, const float* __restrict__ be,
                          const float* __restrict__ Wn, const float* __restrict__ bn,
                          const float* __restrict__ Wu, const float* __restrict__ bu,
                          float* __restrict__ out, int E, int N) {
    __shared__ float mean_e[D_DIM];
    __shared__ float mean_n[D_DIM];
    __shared__ float u_in[3 * H_DIM];

    const int t = threadIdx.x;
    if (t < D_DIM) {
        float s = 0.0f;
        for (int p = 0; p < NB_E; ++p) s += ws[(long)p * D_DIM + t];
        mean_e[t] = s * (1.0f / (float)E);
    } else {
        const int c = t - D_DIM;
        float s = 0.0f;
        for (int p = 0; p < NB_N; ++p) s += ws[(long)(NB_E + p) * D_DIM + c];
        mean_n[c] = s * (1.0f / (float)N);
    }
    __syncthreads();

    const int lane = threadIdx.x & 31;
    const int wave = threadIdx.x >> 5;
    const int hi   = lane >> 4;
    const int col  = (wave << 4) + (lane & 15);

    // Phase 1: two GEMVs in one K loop (independent WMMAs -> ILP)
    v8f accE = {}, accN = {};
    for (int k = 0; k < D_DIM; k += 4) {
        const int ka = k + 2 * hi;
        v2f aE; aE.x = mean_e[ka]; aE.y = mean_e[ka + 1];
        v2f aN; aN.x = mean_n[ka]; aN.y = mean_n[ka + 1];
        v2f bE; bE.x = We[ka * H_DIM + col]; bE.y = We[(ka + 1) * H_DIM + col];
        v2f bN; bN.x = Wn[ka * H_DIM + col]; bN.y = Wn[(ka + 1) * H_DIM + col];
        accE = __builtin_amdgcn_wmma_f32_16x16x4_f32(
            false, aE, false, bE, (short)0, accE, false, false);
        accN = __builtin_amdgcn_wmma_f32_16x16x4_f32(
            false, aN, false, bN, (short)0, accN, false, false);
    }
    if (lane < 16) {
        u_in[col]         = accE[0] + be[col];
        u_in[H_DIM + col] = accN[0] + bn[col];
    }
    if (t < G_DIM) u_in[2 * H_DIM + t] = global_attr[t];
    __syncthreads();

    // Phase 2: out = u_in(1x768) @ Wu(768x256) + bu
    v8f accU = {};
    for (int k = 0; k < 3 * H_DIM; k += 4) {
        const int ka = k + 2 * hi;
        v2f a;  a.x  = u_in[ka];             a.y  = u_in[ka + 1];
        v2f bm; bm.x = Wu[ka * G_DIM + col]; bm.y = Wu[(ka + 1) * G_DIM + col];
        accU = __builtin_amdgcn_wmma_f32_16x16x4_f32(
            false, a, false, bm, (short)0, accU, false, false);
    }
    if (lane < 16) out[col] = accU[0] + bu[col];
}

extern "C" void kernel_launch(void* const* d_in, const int* in_sizes, int n_in,
                              void* d_out, int out_size, void* d_ws, size_t ws_size,
                              hipStream_t stream) {
    const float* edge = (const float*)d_in[0];
    const float* node = (const float*)d_in[1];
    const float* ga   = (const float*)d_in[2];
    const float* We   = (const float*)d_in[3];
    const float* be   = (const float*)d_in[4];
    const float* Wn   = (const float*)d_in[5];
    const float* bn   = (const float*)d_in[6];
    const float* Wu   = (const float*)d_in[7];
    const float* bu   = (const float*)d_in[8];
    float* out = (float*)d_out;
    float* ws  = (float*)d_ws;   // (NB_E + NB_N) * 256 floats = 576 KB partials

    const int E = in_sizes[0] / D_DIM;   // 800000
    const int N = in_sizes[1] / D_DIM;   // 50000

    colsum_wmma_kernel<<<NB_E + NB_N, THREADS, 0, stream>>>(edge, node, ws, E, N);
    global_update_kernel<<<1, THREADS, 0, stream>>>(ws, ga, We, be, Wn, bn, Wu, bu,
                                                    out, E, N);
}